// GNNClassifier_56521769615598
// MI455X (gfx1250) — compile-verified
//
#include <hip/hip_runtime.h>
#include <math.h>

typedef __attribute__((ext_vector_type(2))) float v2f;
typedef __attribute__((ext_vector_type(8))) float v8f;

// ---------------------------------------------------------------- degree prep
__global__ void k_deg_init(unsigned* __restrict__ deg, int N) {
    int n = blockIdx.x * blockDim.x + threadIdx.x;
    if (n < N) deg[n] = 1u;  // self loop
}

__global__ void k_deg_count(const int* __restrict__ dst, unsigned* __restrict__ deg, int E) {
    int e = blockIdx.x * blockDim.x + threadIdx.x;
    if (e < E) atomicAdd(&deg[dst[e]], 1u);
}

__global__ void k_dinv(const unsigned* __restrict__ deg, float* __restrict__ dinv, int N) {
    int n = blockIdx.x * blockDim.x + threadIdx.x;
    if (n < N) dinv[n] = rsqrtf((float)deg[n]);  // deg >= 1 always (self loops)
}

// ------------------------------------------------- WMMA GEMM: g = (A@W)*dinv
// A: [M,K] row-major, W: [K,64] row-major, one wave -> 16 rows x 64 cols.
// Writes the pre-scaled messages g and initializes acc = g (self-loop term).
__global__ void k_gemm_scale(const float* __restrict__ A, const float* __restrict__ W,
                             const float* __restrict__ dinv,
                             float* __restrict__ g, float* __restrict__ acc,
                             int M, int K) {
    int wave = threadIdx.x >> 5;
    int lane = threadIdx.x & 31;
    int tile = blockIdx.x * (blockDim.x >> 5) + wave;
    int m0 = tile * 16;
    if (m0 >= M) return;  // wave-uniform

    int half = lane >> 4;   // 0: K pair {0,1}, 1: K pair {2,3}
    int l16  = lane & 15;

    const float* arow = A + (size_t)(m0 + l16) * K;
    v8f c0 = {}, c1 = {}, c2 = {}, c3 = {};

    for (int kk = 0; kk < K; kk += 4) {
        int ka = kk + 2 * half;
        // A fragment (16x4 f32): VGPR0=K(ka), VGPR1=K(ka+1) for this lane-half
        v2f a;
        a.x = arow[ka];
        a.y = arow[ka + 1];
        // B fragments (4x16 f32): lane = N column, VGPR0=K(ka), VGPR1=K(ka+1)
        const float* w0 = W + (size_t)ka * 64 + l16;
        const float* w1 = w0 + 64;
        v2f b0, b1, b2, b3;
        b0.x = w0[0];  b0.y = w1[0];
        b1.x = w0[16]; b1.y = w1[16];
        b2.x = w0[32]; b2.y = w1[32];
        b3.x = w0[48]; b3.y = w1[48];
        c0 = __builtin_amdgcn_wmma_f32_16x16x4_f32(false, a, false, b0, (short)0, c0, false, false);
        c1 = __builtin_amdgcn_wmma_f32_16x16x4_f32(false, a, false, b1, (short)0, c1, false, false);
        c2 = __builtin_amdgcn_wmma_f32_16x16x4_f32(false, a, false, b2, (short)0, c2, false, false);
        c3 = __builtin_amdgcn_wmma_f32_16x16x4_f32(false, a, false, b3, (short)0, c3, false, false);
    }

    // D layout: VGPR v holds row (m0 + v + 8*half), column = l16 (+tile offs)
    int rbase = m0 + 8 * half;
    for (int v = 0; v < 8; ++v) {
        int row = rbase + v;
        float s = dinv[row];
        size_t o = (size_t)row * 64 + l16;
        float x0 = c0[v] * s, x1 = c1[v] * s, x2 = c2[v] * s, x3 = c3[v] * s;
        g[o]        = x0;  g[o + 16]   = x1;  g[o + 32]   = x2;  g[o + 48]   = x3;
        acc[o]      = x0;  acc[o + 16] = x1;  acc[o + 32] = x2;  acc[o + 48] = x3;
    }
}

// ---------------------------------------------- edge scatter-add (mem bound)
// thread = (edge, feature); a wave covers 32 consecutive feats of one edge.
__global__ void k_edge_agg(const int* __restrict__ src, const int* __restrict__ dst,
                           const float* __restrict__ g, float* __restrict__ acc, int E) {
    long long tid = (long long)blockIdx.x * blockDim.x + threadIdx.x;
    if (tid >= (long long)E * 64) return;
    int e = (int)(tid >> 6);
    int f = (int)(tid & 63);
    float val = g[(size_t)src[e] * 64 + f];
    unsafeAtomicAdd(&acc[(size_t)dst[e] * 64 + f], val);  // global_atomic_add_f32
}

// ----------------------------------------- finalize layer: dinv*acc + b, relu
__global__ void k_finalize_relu(const float* __restrict__ acc, const float* __restrict__ dinv,
                                const float* __restrict__ b, float* __restrict__ out, int N) {
    long long tid = (long long)blockIdx.x * blockDim.x + threadIdx.x;
    if (tid >= (long long)N * 64) return;
    int n = (int)(tid >> 6);
    int f = (int)(tid & 63);
    float v = dinv[n] * acc[tid] + b[f];
    out[tid] = v > 0.0f ? v : 0.0f;
}

// --------------------- final: (dinv*acc2 + b2) @ Wfc + bfc -> sigmoid, [N,1]
__global__ void k_final(const float* __restrict__ acc2, const float* __restrict__ dinv,
                        const float* __restrict__ b2, const float* __restrict__ Wfc,
                        const float* __restrict__ bfc, float* __restrict__ out, int N) {
    int wave = threadIdx.x >> 5;
    int lane = threadIdx.x & 31;
    int n = blockIdx.x * (blockDim.x >> 5) + wave;
    if (n >= N) return;
    float di = dinv[n];
    size_t o = (size_t)n * 64;
    float v0 = di * acc2[o + lane]      + b2[lane];
    float v1 = di * acc2[o + lane + 32] + b2[lane + 32];
    float p = v0 * Wfc[lane] + v1 * Wfc[lane + 32];
    #pragma unroll
    for (int off = 16; off > 0; off >>= 1) p += __shfl_xor(p, off, 32);
    if (lane == 0) {
        float z = p + bfc[0];
        out[n] = 1.0f / (1.0f + expf(-z));
    }
}

extern "C" void kernel_launch(void* const* d_in, const int* in_sizes, int n_in,
                              void* d_out, int out_size, void* d_ws, size_t ws_size,
                              hipStream_t stream) {
    const float* x   = (const float*)d_in[0];
    const int*   ei  = (const int*)d_in[1];
    const float* W1  = (const float*)d_in[2];
    const float* b1  = (const float*)d_in[3];
    const float* W2  = (const float*)d_in[4];
    const float* b2  = (const float*)d_in[5];
    const float* Wfc = (const float*)d_in[6];
    const float* bfc = (const float*)d_in[7];

    const int E = in_sizes[1] / 2;     // 800000
    const int N = in_sizes[0] / 128;   // 50000
    const int* src = ei;
    const int* dst = ei + E;

    // workspace carve-out
    char* ws = (char*)d_ws;
    auto alignup = [](size_t v) { return (v + 255) & ~(size_t)255; };
    size_t off = 0;
    unsigned* deg  = (unsigned*)(ws + off); off += alignup((size_t)N * 4);
    float*    dinv = (float*)(ws + off);    off += alignup((size_t)N * 4);
    float*    gbuf = (float*)(ws + off);    off += alignup((size_t)N * 64 * 4); // g1 then g2
    float*    abuf = (float*)(ws + off);    off += alignup((size_t)N * 64 * 4); // acc1 then acc2
    float*    rbuf = (float*)(ws + off);    off += alignup((size_t)N * 64 * 4); // relu(layer1)
    (void)ws_size; (void)n_in; (void)out_size;

    const int T = 256;
    const long long ework = (long long)E * 64;
    const long long nwork = (long long)N * 64;
    const int tiles   = (N + 15) / 16;
    const int gblocks = (tiles + 7) / 8;   // 8 waves per block

    k_deg_init <<<(N + T - 1) / T, T, 0, stream>>>(deg, N);
    k_deg_count<<<(E + T - 1) / T, T, 0, stream>>>(dst, deg, E);
    k_dinv     <<<(N + T - 1) / T, T, 0, stream>>>(deg, dinv, N);

    // layer 1: g1 = (x@W1)*dinv ; acc1 = g1 ; scatter ; relu(dinv*acc1 + b1)
    k_gemm_scale   <<<gblocks, T, 0, stream>>>(x, W1, dinv, gbuf, abuf, N, 128);
    k_edge_agg     <<<(int)((ework + T - 1) / T), T, 0, stream>>>(src, dst, gbuf, abuf, E);
    k_finalize_relu<<<(int)((nwork + T - 1) / T), T, 0, stream>>>(abuf, dinv, b1, rbuf, N);

    // layer 2: g2 = (h1@W2)*dinv ; acc2 = g2 ; scatter ; fused FC + sigmoid
    k_gemm_scale<<<gblocks, T, 0, stream>>>(rbuf, W2, dinv, gbuf, abuf, N, 64);
    k_edge_agg  <<<(int)((ework + T - 1) / T), T, 0, stream>>>(src, dst, gbuf, abuf, E);
    k_final     <<<(N + 7) / 8, T, 0, stream>>>(abuf, dinv, b2, Wfc, bfc, (float*)d_out, N);
}